// TreeEnsemble_35424890257688
// MI455X (gfx1250) — compile-verified
//
#include <hip/hip_runtime.h>
#include <hip/hip_bf16.h>
#include <cstdint>

// ---------------- problem constants (match reference) ----------------
constexpr int N_BATCH   = 50000;
constexpr int N_FEATURE = 128;
constexpr int N_TREE    = 200;    // even: processed in pairs
constexpr int N_NODE    = 1023;   // 2^(9+1)-1
constexpr int NODE_PAD  = 1024;   // padded per-tree slab (16 KB of uint4)
constexpr int TDEPTH    = 9;
constexpr int N_OUTPUT  = 8;

constexpr int BLK = 128;               // 4 waves (wave32)
constexpr int NPT = NODE_PAD / BLK;    // packed nodes per thread when reg-staging

typedef unsigned int u32;

// ---------------- TDM (Tensor Data Mover) helpers ----------------
#if defined(__HIP_DEVICE_COMPILE__) && defined(__has_builtin)
#if __has_builtin(__builtin_amdgcn_tensor_load_to_lds) && __has_builtin(__builtin_amdgcn_s_wait_tensorcnt)
#define USE_TDM 1
#endif
#endif
#ifndef USE_TDM
#define USE_TDM 0
#endif

#if defined(__HIP_DEVICE_COMPILE__)
#if USE_TDM
#pragma message("CDNA5-PATH: tensor_load_to_lds + s_wait_tensorcnt ENABLED (device pass)")
#else
#pragma message("CDNA5-PATH: TDM builtins NOT available; register-staging fallback (device pass)")
#endif
#endif

#if USE_TDM
typedef __attribute__((ext_vector_type(4))) u32 u32x4;
typedef __attribute__((ext_vector_type(8))) int i32x8;
typedef __attribute__((ext_vector_type(4))) int i32x4;

// low 32 bits of a generic pointer to LDS are the byte offset within the
// workgroup's LDS allocation (flat-address aperture layout, ISA 10.2)
__device__ __forceinline__ u32 lds_off(const void* p) {
  return (u32)(uintptr_t)p;
}

// 1-D contiguous DMA: n_dwords (<= 16384) from global -> LDS, async (TENSORcnt)
__device__ __forceinline__ void tdm_load_1d(u32 lds_byte_off, const void* gsrc,
                                            u32 n_dwords) {
  unsigned long long ga = (unsigned long long)(uintptr_t)gsrc;
  u32x4 g0;
  g0.x = 1u;                                           // count=1, user D#
  g0.y = lds_byte_off;                                 // lds_addr
  g0.z = (u32)(ga & 0xFFFFFFFFull);                    // global_addr[31:0]
  g0.w = (u32)((ga >> 32) & 0x1FFFFFFull) | (2u << 30);// addr[56:32] | type=2

  i32x8 g1;
  const u32 td0 = n_dwords;        // tensor_dim0 (elements)
  const u32 tile0 = n_dwords;      // tile_dim0 (16-bit field; <= 65535)
  g1.s0 = (int)(2u << 16);                               // data_size = 4B
  g1.s1 = (int)((td0 & 0xFFFFu) << 16);                  // tensor_dim0[15:0]
  g1.s2 = (int)(((td0 >> 16) & 0xFFFFu) | (1u << 16));   // td0 hi | tensor_dim1=1
  g1.s3 = (int)((tile0 & 0xFFFFu) << 16);                // tile_dim0
  g1.s4 = 0;                                             // tile_dim1/2 unused
  g1.s5 = (int)td0;                                      // tensor_dim0_stride lo
  g1.s6 = 0;
  g1.s7 = 0;

  i32x4 z4 = {0, 0, 0, 0};
#if __has_include(<hip/amd_detail/amd_gfx1250_TDM.h>)
  i32x8 z8 = {0, 0, 0, 0, 0, 0, 0, 0};
  __builtin_amdgcn_tensor_load_to_lds(g0, g1, z4, z4, z8, 0);   // clang-23 form
#else
  __builtin_amdgcn_tensor_load_to_lds(g0, g1, z4, z4, 0);       // ROCm 7.2 form
#endif
}

__device__ __forceinline__ void tdm_wait() {
  __builtin_amdgcn_s_wait_tensorcnt(0);
}
#else
__device__ __forceinline__ u32  lds_off(const void*) { return 0u; }
__device__ __forceinline__ void tdm_load_1d(u32, const void*, u32) {}
__device__ __forceinline__ void tdm_wait() {}
#endif

// ---------------- pack kernel: SoA -> AoS {feat, thr, cl, cr} ----------------
__global__ void tree_pack_kernel(const int* __restrict__ feat,
                                 const float* __restrict__ thr,
                                 const int* __restrict__ cl,
                                 const int* __restrict__ cr,
                                 uint4* __restrict__ packed) {
  int i = blockIdx.x * blockDim.x + threadIdx.x;
  if (i >= N_TREE * NODE_PAD) return;
  int t = i / NODE_PAD;
  int n = i - t * NODE_PAD;
  uint4 v;
  if (n < N_NODE) {
    int s = t * N_NODE + n;
    v = make_uint4((u32)feat[s], __float_as_uint(thr[s]), (u32)cl[s], (u32)cr[s]);
  } else {
    v = make_uint4((u32)(-2), 0u, (u32)(-1), (u32)(-1));  // padding: leaf-like
  }
  packed[i] = v;
}

// ---------------- per-tree register staging (fallback path) ----------------
template <bool PACKED>
__device__ __forceinline__ void stage_to_regs(int t, int lid, uint4 (&r)[NPT],
                                              const uint4* __restrict__ packed,
                                              const int* __restrict__ gf,
                                              const float* __restrict__ gth,
                                              const int* __restrict__ gcl,
                                              const int* __restrict__ gcr) {
#pragma unroll
  for (int k = 0; k < NPT; ++k) {
    int i = lid + k * BLK;
    if (PACKED) {
      r[k] = packed[(size_t)t * NODE_PAD + i];
    } else if (i < N_NODE) {
      int s = t * N_NODE + i;
      r[k] = make_uint4((u32)gf[s], __float_as_uint(gth[s]), (u32)gcl[s],
                        (u32)gcr[s]);
    } else {
      r[k] = make_uint4((u32)(-2), 0u, (u32)(-1), (u32)(-1));
    }
  }
}

// ---------------- main kernel (processes trees in pairs: 2-way ILP) --------
template <bool PACKED>
__global__ __launch_bounds__(BLK) void tree_ensemble_kernel(
    const float* __restrict__ x, const uint4* __restrict__ packed,
    const int* __restrict__ gf, const float* __restrict__ gth,
    const int* __restrict__ gcl, const int* __restrict__ gcr,
    const float* __restrict__ value, float* __restrict__ out) {
  extern __shared__ float xs[];                 // BLK * N_FEATURE floats (64 KB)
  __shared__ uint4 ndbuf[2][2][NODE_PAD];       // [pair-buffer][tree-in-pair] 64 KB

  constexpr bool TDMPATH = PACKED && (USE_TDM != 0);
  constexpr int NPAIR = N_TREE / 2;             // 100

  const int lid = threadIdx.x;
  const int blockStart = blockIdx.x * BLK;
  const int g = blockStart + lid;
  const bool valid = g < N_BATCH;
  const int rows = min(BLK, N_BATCH - blockStart);

  // ---- prologue: stage x tile + tree pair 0 ----
  if constexpr (TDMPATH) {
    if (lid < 32) {  // wave 0 issues the DMAs; TDM ignores EXEC, per-wave op
      tdm_load_1d(lds_off(&xs[0]), x + (size_t)blockStart * N_FEATURE,
                  (u32)(rows * N_FEATURE));
      tdm_load_1d(lds_off(&ndbuf[0][0][0]), packed, NODE_PAD * 4);
      tdm_load_1d(lds_off(&ndbuf[0][1][0]), packed + (size_t)NODE_PAD,
                  NODE_PAD * 4);
      tdm_wait();
    }
  } else {
    // cooperative vectorized x-tile copy
    const float4* src = (const float4*)(x + (size_t)blockStart * N_FEATURE);
    float4* dst = (float4*)xs;
    const int n4 = rows * (N_FEATURE / 4);
    for (int i = lid; i < n4; i += BLK) dst[i] = src[i];
    uint4 ra[NPT], rb[NPT];
    stage_to_regs<PACKED>(0, lid, ra, packed, gf, gth, gcl, gcr);
    stage_to_regs<PACKED>(1, lid, rb, packed, gf, gth, gcl, gcr);
#pragma unroll
    for (int k = 0; k < NPT; ++k) {
      ndbuf[0][0][lid + k * BLK] = ra[k];
      ndbuf[0][1][lid + k * BLK] = rb[k];
    }
  }
  __syncthreads();

  const float* xrow = &xs[(valid ? lid : 0) * N_FEATURE];

  float a0 = 0.f, a1 = 0.f, a2 = 0.f, a3 = 0.f;
  float a4 = 0.f, a5 = 0.f, a6 = 0.f, a7 = 0.f;

  for (int p = 0; p < NPAIR; ++p) {
    const int pb = p & 1;
    const int nb2 = pb ^ 1;
    const int t0 = 2 * p;

    // kick off staging of next tree pair (overlaps with traversal)
    uint4 ra[NPT], rb[NPT];
    if (p + 1 < NPAIR) {
      if constexpr (TDMPATH) {
        if (lid < 32) {
          tdm_load_1d(lds_off(&ndbuf[nb2][0][0]),
                      packed + (size_t)(t0 + 2) * NODE_PAD, NODE_PAD * 4);
          tdm_load_1d(lds_off(&ndbuf[nb2][1][0]),
                      packed + (size_t)(t0 + 3) * NODE_PAD, NODE_PAD * 4);
        }
      } else {
        stage_to_regs<PACKED>(t0 + 2, lid, ra, packed, gf, gth, gcl, gcr);
        stage_to_regs<PACKED>(t0 + 3, lid, rb, packed, gf, gth, gcl, gcr);
      }
      // warm next pair's leaf-value slabs (2 x 32 KB) into near caches
      const char* pv0 = (const char*)(value + (size_t)(t0 + 2) * N_NODE * N_OUTPUT);
      const char* pv1 = (const char*)(value + (size_t)(t0 + 3) * N_NODE * N_OUTPUT);
      __builtin_prefetch(pv0 + (size_t)lid * 256, 0, 1);
      __builtin_prefetch(pv0 + (size_t)lid * 256 + 128, 0, 1);
      __builtin_prefetch(pv1 + (size_t)lid * 256, 0, 1);
      __builtin_prefetch(pv1 + (size_t)lid * 256 + 128, 0, 1);
    }

    // ---- traverse trees t0 and t0+1 with interleaved independent chains ----
    const uint4* nbA = ndbuf[pb][0];
    const uint4* nbB = ndbuf[pb][1];
    int nA = 0, nB = 0;
#pragma unroll
    for (int d = 0; d < TDEPTH; ++d) {
      uint4 da = nbA[nA];                        // {feat, thr, cl, cr}
      uint4 db = nbB[nB];
      int fA = (int)da.x;
      int fB = (int)db.x;
      float xA = (fA < 0) ? 0.0f : xrow[fA];     // padded-column semantics
      float xB = (fB < 0) ? 0.0f : xrow[fB];
      float tA = __uint_as_float(da.y);
      float tB = __uint_as_float(db.y);
      int cA = (xA <= tA) ? (int)da.z : (int)da.w;
      int cB = (xB <= tB) ? (int)db.z : (int)db.w;
      nA = (cA > -1) ? cA : nA;                  // stay put at leaf marker
      nB = (cB > -1) ? cB : nB;
    }

    // ---- accumulate leaf values (four float4 gathers, 32B aligned) ----
    const float4* vpA =
        (const float4*)(value + ((size_t)t0 * N_NODE + nA) * N_OUTPUT);
    const float4* vpB =
        (const float4*)(value + ((size_t)(t0 + 1) * N_NODE + nB) * N_OUTPUT);
    float4 va0 = vpA[0], va1 = vpA[1];
    float4 vb0 = vpB[0], vb1 = vpB[1];
    a0 += va0.x + vb0.x; a1 += va0.y + vb0.y;
    a2 += va0.z + vb0.z; a3 += va0.w + vb0.w;
    a4 += va1.x + vb1.x; a5 += va1.y + vb1.y;
    a6 += va1.z + vb1.z; a7 += va1.w + vb1.w;

    // ---- finish staging of next pair, publish buffers ----
    if (p + 1 < NPAIR) {
      if constexpr (TDMPATH) {
        if (lid < 32) tdm_wait();
      } else {
#pragma unroll
        for (int k = 0; k < NPT; ++k) {
          ndbuf[nb2][0][lid + k * BLK] = ra[k];
          ndbuf[nb2][1][lid + k * BLK] = rb[k];
        }
      }
    }
    __syncthreads();
  }

  if (valid) {
    const float s = 1.0f / (float)N_TREE;
    float4* o = (float4*)(out + (size_t)g * N_OUTPUT);
    o[0] = make_float4(a0 * s, a1 * s, a2 * s, a3 * s);
    o[1] = make_float4(a4 * s, a5 * s, a6 * s, a7 * s);
  }
}

// ---------------- host launcher ----------------
extern "C" void kernel_launch(void* const* d_in, const int* in_sizes, int n_in,
                              void* d_out, int out_size, void* d_ws,
                              size_t ws_size, hipStream_t stream) {
  (void)in_sizes; (void)n_in; (void)out_size;
  const float* x        = (const float*)d_in[0];
  const int*   feature  = (const int*)d_in[1];
  const float* threshold= (const float*)d_in[2];
  const int*   cleft    = (const int*)d_in[3];
  const int*   cright   = (const int*)d_in[4];
  const float* value    = (const float*)d_in[5];
  float*       out      = (float*)d_out;

  const int grid = (N_BATCH + BLK - 1) / BLK;             // 391 blocks
  const size_t xsBytes = (size_t)BLK * N_FEATURE * sizeof(float);  // 64 KB

  const size_t packedBytes = (size_t)N_TREE * NODE_PAD * sizeof(uint4);
  const bool usePacked = ws_size >= packedBytes;

  if (usePacked) {
    uint4* packed = (uint4*)d_ws;
    const int packN = N_TREE * NODE_PAD;
    tree_pack_kernel<<<(packN + 255) / 256, 256, 0, stream>>>(
        feature, threshold, cleft, cright, packed);
    tree_ensemble_kernel<true><<<grid, BLK, xsBytes, stream>>>(
        x, packed, feature, threshold, cleft, cright, value, out);
  } else {
    tree_ensemble_kernel<false><<<grid, BLK, xsBytes, stream>>>(
        x, nullptr, feature, threshold, cleft, cright, value, out);
  }
}